// Correlation_30992484008482
// MI455X (gfx1250) — compile-verified
//
#include <hip/hip_runtime.h>
#include <stdint.h>

// out[i,j,b,c,y,x] = leaky_relu(ref[b,c,y,x] * tgt[b,c,y+j-4,x+i-4], 0.1)
// ref/tgt: (4,64,64,128) f32.  out: (9,9,4,64,64,128) f32.
// Pure streaming-store problem (~680MB out vs 0.34 GFLOP) -> stage inputs with
// CDNA5 async global->LDS DMA, write output with non-temporal B128 stores.

typedef __attribute__((ext_vector_type(4))) float v4f;

__device__ __forceinline__ float leaky(float x) {
  return x > 0.0f ? x : 0.1f * x;
}

__global__ __launch_bounds__(128, 4) void corr_async_kernel(
    const float* __restrict__ ref,
    const float* __restrict__ tgt,
    float* __restrict__ out)
{
  // 4 ref rows (128 floats each) + 12 padded tgt rows (144-float stride:
  // [0..7] zero apron | [8..135] data | [136..143] zero apron)
  __shared__ __align__(16) float lds_ref[4 * 128];
  __shared__ __align__(16) float lds_tgt[12 * 144];

  const int t    = threadIdx.x;      // 0..127
  const int lane = t & 31;           // wave32 lane
  const int w    = t >> 5;           // wave id 0..3 (uniform per wave)
  const int bc   = blockIdx.x >> 4;  // flattened (b*64 + c), 0..255
  const int y0   = (blockIdx.x & 15) << 2;  // first of 4 output rows

  const float* refbase = ref + (size_t)bc * 8192;  // 64*128 plane
  const float* tgtbase = tgt + (size_t)bc * 8192;

  // Zero the left/right aprons of every padded tgt row: indices [0..7] and
  // [136..143]. These are disjoint from the async-load destination region
  // [8..135], so there is no DS-vs-async ordering hazard.
  for (int k = t; k < 12 * 16; k += 128) {
    const int jt = k >> 4, e = k & 15;
    lds_tgt[jt * 144 + (e < 8 ? e : 128 + e)] = 0.0f;
  }

  const unsigned ldsr = (unsigned)(uintptr_t)(void*)&lds_ref[0];
  const unsigned ldst = (unsigned)(uintptr_t)(void*)&lds_tgt[0];

  // 16 rows total (4 ref + 12 tgt), one 512B row per wave per iteration:
  // 32 lanes x b128 = one full row per async instruction.
  #pragma unroll
  for (int k = 0; k < 4; ++k) {
    const int r = k * 4 + w;  // uniform per wave -> full EXEC inside branches
    if (r < 4) {
      const unsigned long long g =
          (unsigned long long)(uintptr_t)(refbase + (y0 + r) * 128 + lane * 4);
      const unsigned l = ldsr + (unsigned)(r * 512 + lane * 16);
      asm volatile("global_load_async_to_lds_b128 %0, %1, off"
                   :: "v"(l), "v"(g) : "memory");
    } else {
      const int jt = r - 4;         // padded tgt row 0..11
      const int yj = y0 + jt - 4;   // source row in tgt plane
      if (0 <= yj && yj < 64) {
        const unsigned long long g =
            (unsigned long long)(uintptr_t)(tgtbase + yj * 128 + lane * 4);
        const unsigned l = ldst + (unsigned)(jt * 576 + 32 + lane * 16);
        asm volatile("global_load_async_to_lds_b128 %0, %1, off"
                     :: "v"(l), "v"(g) : "memory");
      } else {
        // Out-of-bounds source row: zero-fill (32 lanes x 16B = 512B row).
        const v4f z = {0.f, 0.f, 0.f, 0.f};
        *(v4f*)&lds_tgt[jt * 144 + 8 + lane * 4] = z;
      }
    }
  }

  asm volatile("s_wait_asynccnt 0" ::: "memory");  // this wave's DMAs done
  __syncthreads();                                  // all waves' data visible

  // Compute: thread owns output row (y0 + w), x in [x4, x4+4).
  const int ysub = w;
  const int x4   = lane * 4;
  const v4f rv = *(const v4f*)&lds_ref[ysub * 128 + x4];
  float* const outp =
      out + (size_t)bc * 8192 + (size_t)(y0 + ysub) * 128 + x4;

  for (int j = 0; j < 9; ++j) {
    // tgt padded row for (y0+ysub)+j-4 is index ysub+j in [0,11].
    const int jt = ysub + j;
    // Sliding window: element (x+i-4) lives at padded index 8+(x+i-4)
    // = x4+comp+i+4; for comp in [0,3], i in [0,8] that spans
    // padded indices x4+4 .. x4+15 (12 floats).
    float tr[12];
    #pragma unroll
    for (int k2 = 0; k2 < 12; ++k2)
      tr[k2] = lds_tgt[jt * 144 + 4 + x4 + k2];

    #pragma unroll
    for (int i = 0; i < 9; ++i) {
      v4f p;
      p.x = leaky(rv.x * tr[i + 0]);
      p.y = leaky(rv.y * tr[i + 1]);
      p.z = leaky(rv.z * tr[i + 2]);
      p.w = leaky(rv.w * tr[i + 3]);
      // (i,j) plane stride = 4*64*64*128 elements; streaming store, bypass
      // cache retention (output is 679MB >> 192MB L2).
      __builtin_nontemporal_store(
          p, (v4f*)(outp + (size_t)(i * 9 + j) * 2097152));
    }
  }
}

extern "C" void kernel_launch(void* const* d_in, const int* in_sizes, int n_in,
                              void* d_out, int out_size, void* d_ws, size_t ws_size,
                              hipStream_t stream) {
  (void)in_sizes; (void)n_in; (void)out_size; (void)d_ws; (void)ws_size;
  const float* ref = (const float*)d_in[0];
  const float* tgt = (const float*)d_in[1];
  float* out = (float*)d_out;
  // 256 (b,c) planes x 16 y-chunks of 4 rows = 4096 blocks, 128 threads each.
  hipLaunchKernelGGL(corr_async_kernel, dim3(4096), dim3(128), 0, stream,
                     ref, tgt, out);
}